// LocallyConnected2d_77515569758640
// MI455X (gfx1250) — compile-verified
//
#include <hip/hip_runtime.h>

typedef __attribute__((ext_vector_type(2))) float v2f;
typedef __attribute__((ext_vector_type(8))) float v8f;

// Problem constants: B=16, inC=32, outC=32, H=W=64, 3x3 stride1 pad1
#define NB    16
#define CIN   32
#define COUT  32
#define HW    64
#define HW2   4096        // 64*64
#define KTOT  288         // CIN * 9

// Steady-state GEMM core: K consumed in chunks of 36 (4 channels x 9 taps
// = 9 wmma steps of K=4). All unfold addressing is precomputed in voff/msk;
// the chunk advance is one uniform pointer bump (A) + one VGPR pointer bump (B).
template <bool MASKED>
__device__ __forceinline__ void lc2d_core(const float* __restrict__ xb,
                                          const float* __restrict__ p0,
                                          const float* __restrict__ p1,
                                          const int (&voff)[9][2],
                                          const float (&msk)[9][2],
                                          v8f& acc0, v8f& acc1)
{
  for (int c0 = 0; c0 < CIN; c0 += 4) {
    #pragma unroll
    for (int s = 0; s < 9; ++s) {
      // B: two contiguous k elements per N-tile (weight k-major) -> b64 loads
      v2f b0 = *(const v2f*)(p0 + s * 4);
      v2f b1 = *(const v2f*)(p1 + s * 4);
      // A: uniform base + per-lane precomputed 32-bit offset (GVS form)
      float a0 = xb[voff[s][0]];
      float a1 = xb[voff[s][1]];
      if (MASKED) { a0 *= msk[s][0]; a1 *= msk[s][1]; }
      v2f a; a.x = a0; a.y = a1;
      acc0 = __builtin_amdgcn_wmma_f32_16x16x4_f32(false, a, false, b0,
                                                   (short)0, acc0, false, false);
      acc1 = __builtin_amdgcn_wmma_f32_16x16x4_f32(false, a, false, b1,
                                                   (short)0, acc1, false, false);
    }
    xb += 4 * HW2;          // uniform (SGPR) advance: next 4 channels
    p0 += 36;               // per-lane weight row advance
    p1 += 36;
  }
}

// One wave32 per spatial position. M = batch = 16, N = outC = 32 (two 16x16
// accumulators), K = 288 via v_wmma_f32_16x16x4_f32.
__global__ __launch_bounds__(256) void lc2d_wmma_f32(
    const float* __restrict__ x,     // [16, 32, 64, 64]
    const float* __restrict__ wgt,   // [64, 64, 32, 288]
    const float* __restrict__ bias,  // [32, 64, 64]
    float* __restrict__ out)         // [16, 32, 64, 64]
{
  const int lane = threadIdx.x & 31;
  const int wave = threadIdx.x >> 5;
  const int pos  = blockIdx.x * 8 + wave;   // 0..4095
  const int h    = pos >> 6;
  const int w    = pos & 63;

  const int lo = lane & 15;   // M row (A) / N col (B,C,D)
  const int hi = lane >> 4;   // K sub-pair selector for A/B operands

  // Per-lane weight row pointers (o = lo and o = lo+16), pre-biased by 2*hi
  const float* p0 = wgt + (size_t)pos * (COUT * KTOT) + (size_t)lo * KTOT + 2 * hi;
  const float* p1 = p0 + 16 * KTOT;

  // Precompute unfold offsets/masks: identical pattern for every 4-channel
  // chunk. kk = position within chunk's 36 reduction slots for this lane.
  int   voff[9][2];
  float msk[9][2];
  #pragma unroll
  for (int s = 0; s < 9; ++s) {
    #pragma unroll
    for (int j = 0; j < 2; ++j) {
      const int kk = s * 4 + 2 * hi + j;     // 0..35
      const int cc = kk / 9;                 // channel delta within chunk
      const int t  = kk - 9 * cc;            // tap 0..8
      const int kh = t / 3;
      const int kw = t - 3 * kh;
      const int ih = h + kh - 1;
      const int iw = w + kw - 1;
      const bool ok = ((unsigned)ih < (unsigned)HW) & ((unsigned)iw < (unsigned)HW);
      voff[s][j] = (lo * CIN + cc) * HW2 + (ok ? (ih * HW + iw) : 0);
      msk[s][j]  = ok ? 1.0f : 0.0f;
    }
  }

  v8f acc0 = {};
  v8f acc1 = {};

  // Wave-uniform interior/border split; force a scalar branch.
  const int interior = ((h > 0) & (h < HW - 1) & (w > 0) & (w < HW - 1)) ? 1 : 0;
  if (__builtin_amdgcn_readfirstlane(interior)) {
    lc2d_core<false>(x, p0, p1, voff, msk, acc0, acc1);
  } else {
    lc2d_core<true>(x, p0, p1, voff, msk, acc0, acc1);
  }

  const float bias0 = bias[(size_t)lo        * HW2 + pos];
  const float bias1 = bias[(size_t)(lo + 16) * HW2 + pos];

  // C/D layout: VGPR r -> M = r (lanes 0-15) / M = 8+r (lanes 16-31), N = lo
  #pragma unroll
  for (int r = 0; r < 8; ++r) {
    const int b = r + 8 * hi;
    out[(((size_t)b * COUT + lo)      * HW + h) * HW + w] = acc0[r] + bias0;
    out[(((size_t)b * COUT + lo + 16) * HW + h) * HW + w] = acc1[r] + bias1;
  }
}

extern "C" void kernel_launch(void* const* d_in, const int* in_sizes, int n_in,
                              void* d_out, int out_size, void* d_ws, size_t ws_size,
                              hipStream_t stream) {
  (void)in_sizes; (void)n_in; (void)out_size; (void)d_ws; (void)ws_size;
  const float* x    = (const float*)d_in[0];
  const float* wgt  = (const float*)d_in[1];
  const float* bias = (const float*)d_in[2];
  float* out = (float*)d_out;

  // 4096 positions, 8 waves (positions) per 256-thread block -> 512 blocks
  dim3 grid(512), block(256);
  lc2d_wmma_f32<<<grid, block, 0, stream>>>(x, wgt, bias, out);
}